// GNN_73083163508964
// MI455X (gfx1250) — compile-verified
//
#include <hip/hip_runtime.h>

typedef float v2f __attribute__((ext_vector_type(2)));
typedef float v8f __attribute__((ext_vector_type(8)));

#define NB 16    // batch
#define NN 256   // nodes

// ---------------- KA1: per-(b,i) row reductions: moc[i][k], diag[i][k] ----
template<int S_IN>
__global__ void k_rowstats(const float* __restrict__ A, float* __restrict__ moc,
                           float* __restrict__ diag) {
  const int bi = blockIdx.x;          // b*256 + i
  const int b = bi >> 8, i = bi & 255;
  const int lane = threadIdx.x;       // 32 threads
  const float* row = A + (size_t)(b * NN + i) * NN * S_IN;
  if (S_IN == 1) {
    float s = 0.f;
    for (int j = lane; j < NN; j += 32) s += row[j];
    for (int off = 16; off > 0; off >>= 1) s += __shfl_down(s, off, 32);
    float* mo = moc + (size_t)(b * NN + i) * 32;
    float* di = diag + (size_t)(b * NN + i) * 32;
    if (lane == 0) { mo[0] = s * (1.f / NN); di[0] = row[i]; }
    else           { mo[lane] = 0.f;         di[lane] = 0.f; }
  } else {
    float s = 0.f;
    for (int j = 0; j < NN; ++j) s += row[j * S_IN + lane];
    moc [(size_t)(b * NN + i) * 32 + lane] = s * (1.f / NN);
    diag[(size_t)(b * NN + i) * 32 + lane] = row[i * S_IN + lane];
  }
}

// ---------------- KA1b: per-batch means + broadcast constants -------------
__global__ void k_means_consts(const float* __restrict__ moc, const float* __restrict__ diag,
                               const float* __restrict__ Xin, int x_in, int S_in,
                               const float* __restrict__ A_bias, const float* __restrict__ A_l2,
                               const float* __restrict__ A_l4, const float* __restrict__ A_l7,
                               const float* __restrict__ X1_bias, const float* __restrict__ X1_l2,
                               const float* __restrict__ X1_l5, const float* __restrict__ X1_l6,
                               const float* __restrict__ X2_bias, const float* __restrict__ X2_l2,
                               const float* __restrict__ X2_l5, const float* __restrict__ X2_l6,
                               float* __restrict__ baseB, float* __restrict__ c1,
                               float* __restrict__ c2) {
  const int b = blockIdx.x, s = threadIdx.x;   // 32 threads
  float md = 0.f, ma = 0.f, mx = 0.f;
  for (int i = 0; i < NN; ++i) {
    md += diag[(size_t)(b * NN + i) * 32 + s];
    ma += moc [(size_t)(b * NN + i) * 32 + s];
  }
  if (s < x_in)
    for (int i = 0; i < NN; ++i) mx += Xin[(size_t)(b * NN + i) * x_in + s];
  __shared__ float sMD[32], sMA[32], sMX[32];
  sMD[s] = md * (1.f / NN);
  sMA[s] = ma * (1.f / NN);
  sMX[s] = (s < x_in) ? mx * (1.f / NN) : 0.f;
  __syncthreads();
  float bb = A_bias[s], v1 = X1_bias[s], v2 = X2_bias[s];
  for (int k = 0; k < S_in; ++k) {
    bb += sMA[k] * A_l2 [k * 32 + s] + sMD[k] * A_l4 [k * 32 + s];
    v1 += sMA[k] * X1_l6[k * 32 + s] + sMD[k] * X1_l5[k * 32 + s];
    v2 += sMA[k] * X2_l6[k * 32 + s] + sMD[k] * X2_l5[k * 32 + s];
  }
  for (int c = 0; c < x_in; ++c) {
    bb += sMX[c] * A_l7 [c * 32 + s];
    v1 += sMX[c] * X1_l2[c * 32 + s];
    v2 += sMX[c] * X2_l2[c * 32 + s];
  }
  baseB[b * 32 + s] = bb;
  c1[b * 32 + s] = v1;
  c2[b * 32 + s] = v2;
}

// ---------------- KA2: r[i][s], X1t[i][t], X2tT[t][i] ---------------------
__global__ void k_rowmats(const float* __restrict__ moc, const float* __restrict__ diag,
                          const float* __restrict__ Xin, int x_in, int S_in,
                          const float* __restrict__ A_l3, const float* __restrict__ A_l5,
                          const float* __restrict__ A_l6,
                          const float* __restrict__ X1_l1, const float* __restrict__ X1_l3,
                          const float* __restrict__ X1_l4,
                          const float* __restrict__ X2_l1, const float* __restrict__ X2_l3,
                          const float* __restrict__ X2_l4,
                          const float* __restrict__ c1, const float* __restrict__ c2,
                          float* __restrict__ rArr, float* __restrict__ X1t,
                          float* __restrict__ X2tT) {
  const int b = blockIdx.x >> 5, ib = blockIdx.x & 31;
  const int i = ib * 8 + threadIdx.y, t = threadIdx.x;   // block (32,8)
  const float* mo = moc  + (size_t)(b * NN + i) * 32;
  const float* di = diag + (size_t)(b * NN + i) * 32;
  const float* Xr = Xin  + (size_t)(b * NN + i) * x_in;
  float rv = 0.f, v1 = c1[b * 32 + t], v2 = c2[b * 32 + t];
  for (int k = 0; k < S_in; ++k) {
    float m = mo[k], d = di[k];
    rv += m * A_l3 [k * 32 + t] + d * A_l5 [k * 32 + t];
    v1 += m * X1_l3[k * 32 + t] + d * X1_l4[k * 32 + t];
    v2 += m * X2_l3[k * 32 + t] + d * X2_l4[k * 32 + t];
  }
  for (int c = 0; c < x_in; ++c) {
    float x = Xr[c];
    rv += x * A_l6 [c * 32 + t];
    v1 += x * X1_l1[c * 32 + t];
    v2 += x * X2_l1[c * 32 + t];
  }
  rArr[(size_t)(b * NN + i) * 32 + t] = rv;
  X1t [(size_t)(b * NN + i) * 32 + t] = v1;
  X2tT[(size_t)b * (32 * NN) + t * NN + i] = v2;
}

// ---------------- KB: At[i][j][s] = A@A_l1 + r_i + r_j + base (WMMA) ------
// One wave per (b, i, 16-wide j tile). In-place safe: each (i,j,*) row is
// read and written by exactly this wave; loads complete (feed WMMA) first.
template<int S_IN>
__global__ void k_At(const float* Ain, const float* __restrict__ A_l1,
                     const float* __restrict__ rArr, const float* __restrict__ baseB,
                     float* Aout) {
  const int blk = blockIdx.x;
  const int jt = blk & 15, i = (blk >> 4) & 255, b = blk >> 12;
  const int lane = threadIdx.x, half = lane >> 4, lm = lane & 15;
  const int j0 = jt * 16;
  const float* Arow = Ain + ((size_t)(b * NN + i) * NN + (j0 + lm)) * S_IN;
  v8f d0 = {0,0,0,0,0,0,0,0};
  v8f d1 = {0,0,0,0,0,0,0,0};
  const int KP = (S_IN + 3) & ~3;
  for (int k0 = 0; k0 < KP; k0 += 4) {
    const int ka = k0 + 2 * half;
    v2f a, b0, b1;
    a.x  = (ka     < S_IN) ? Arow[ka]     : 0.f;
    a.y  = (ka + 1 < S_IN) ? Arow[ka + 1] : 0.f;
    b0.x = (ka     < S_IN) ? A_l1[ka * 32 + lm]            : 0.f;
    b0.y = (ka + 1 < S_IN) ? A_l1[(ka + 1) * 32 + lm]      : 0.f;
    b1.x = (ka     < S_IN) ? A_l1[ka * 32 + 16 + lm]       : 0.f;
    b1.y = (ka + 1 < S_IN) ? A_l1[(ka + 1) * 32 + 16 + lm] : 0.f;
    d0 = __builtin_amdgcn_wmma_f32_16x16x4_f32(false, a, false, b0, (short)0, d0, false, false);
    d1 = __builtin_amdgcn_wmma_f32_16x16x4_f32(false, a, false, b1, (short)0, d1, false, false);
  }
  const float* ri = rArr + (size_t)(b * NN + i) * 32;
  const float ri0 = ri[lm], ri1 = ri[16 + lm];
  const float bb0 = baseB[b * 32 + lm], bb1 = baseB[b * 32 + 16 + lm];
  float* Aor = Aout + (size_t)(b * NN + i) * NN * 32;
#pragma unroll
  for (int v = 0; v < 8; ++v) {
    const int j = j0 + v + 8 * half;
    const float* rj = rArr + (size_t)(b * NN + j) * 32;
    Aor[(size_t)j * 32 + lm]      = d0[v] + ri0 + rj[lm]      + bb0;
    Aor[(size_t)j * 32 + 16 + lm] = d1[v] + ri1 + rj[16 + lm] + bb1;
  }
}

// ---------------- KC: Y = At_i^T @ X2t (WMMA), project with out_W ---------
// block = 128 threads (4 waves); each wave owns one (b, i).
__global__ void k_einsum_out(const float* __restrict__ At, const float* __restrict__ X2tT,
                             const float* __restrict__ X1t, const float* __restrict__ outW,
                             const float* __restrict__ outB, float* __restrict__ Xout,
                             int do_relu) {
  __shared__ float Y[4][32][33];   // Y^T stored as [t][s]
  const int w = threadIdx.x >> 5, lane = threadIdx.x & 31;
  const int half = lane >> 4, lm = lane & 15;
  const int b = blockIdx.x >> 6;
  const int i = (blockIdx.x & 63) * 4 + w;
  const float* Ai  = At + (size_t)(b * NN + i) * NN * 32;
  const float* X2T = X2tT + (size_t)b * (32 * NN);
  v8f a00 = {0,0,0,0,0,0,0,0}, a01 = {0,0,0,0,0,0,0,0};
  v8f a10 = {0,0,0,0,0,0,0,0}, a11 = {0,0,0,0,0,0,0,0};
  for (int j0 = 0; j0 < NN; j0 += 4) {
    const int kk = j0 + 2 * half;
    v2f fa0 = *(const v2f*)(X2T + (size_t)lm * NN + kk);
    v2f fa1 = *(const v2f*)(X2T + (size_t)(16 + lm) * NN + kk);
    v2f fb0, fb1;
    fb0.x = Ai[(size_t)kk * 32 + lm];
    fb0.y = Ai[(size_t)(kk + 1) * 32 + lm];
    fb1.x = Ai[(size_t)kk * 32 + 16 + lm];
    fb1.y = Ai[(size_t)(kk + 1) * 32 + 16 + lm];
    a00 = __builtin_amdgcn_wmma_f32_16x16x4_f32(false, fa0, false, fb0, (short)0, a00, false, false);
    a01 = __builtin_amdgcn_wmma_f32_16x16x4_f32(false, fa0, false, fb1, (short)0, a01, false, false);
    a10 = __builtin_amdgcn_wmma_f32_16x16x4_f32(false, fa1, false, fb0, (short)0, a10, false, false);
    a11 = __builtin_amdgcn_wmma_f32_16x16x4_f32(false, fa1, false, fb1, (short)0, a11, false, false);
  }
#pragma unroll
  for (int v = 0; v < 8; ++v) {
    const int m = v + 8 * half;
    Y[w][m][lm]           = a00[v];
    Y[w][m][16 + lm]      = a01[v];
    Y[w][16 + m][lm]      = a10[v];
    Y[w][16 + m][16 + lm] = a11[v];
  }
  __syncthreads();
  // out[i][t] = (1/n) * sum_{s,t2} Y[s][t2] * outW[(s*32+t2)*32 + t] + out_b + X1t
  float acc = 0.f;
  for (int s = 0; s < 32; ++s)
    for (int t2 = 0; t2 < 32; ++t2)
      acc += Y[w][t2][s] * outW[(size_t)(s * 32 + t2) * 32 + lane];
  float val = acc * (1.f / NN) + outB[lane] + X1t[(size_t)(b * NN + i) * 32 + lane];
  if (do_relu) val = fmaxf(val, 0.f);
  Xout[(size_t)(b * NN + i) * 32 + lane] = val;
}

extern "C" void kernel_launch(void* const* d_in, const int* in_sizes, int n_in,
                              void* d_out, int out_size, void* d_ws, size_t ws_size,
                              hipStream_t stream) {
  (void)in_sizes; (void)n_in; (void)out_size; (void)ws_size;
  const float* adj  = (const float*)d_in[0];
  const float* feat = (const float*)d_in[1];

  char* ws = (char*)d_ws;
  const size_t A_BYTES   = (size_t)NB * NN * NN * 32 * sizeof(float);  // 134 MB
  const size_t ROW_BYTES = (size_t)NB * NN * 32 * sizeof(float);       // 512 KB
  float* A0   = (float*)ws;                 ws += A_BYTES;
  float* Xa   = (float*)ws;                 ws += ROW_BYTES;
  float* Xb   = (float*)ws;                 ws += ROW_BYTES;
  float* moc  = (float*)ws;                 ws += ROW_BYTES;
  float* diag = (float*)ws;                 ws += ROW_BYTES;
  float* rArr = (float*)ws;                 ws += ROW_BYTES;
  float* X1t  = (float*)ws;                 ws += ROW_BYTES;
  float* X2tT = (float*)ws;                 ws += ROW_BYTES;
  float* baseB= (float*)ws;                 ws += NB * 32 * sizeof(float);
  float* c1   = (float*)ws;                 ws += NB * 32 * sizeof(float);
  float* c2   = (float*)ws;                 ws += NB * 32 * sizeof(float);

  for (int l = 0; l < 3; ++l) {
    const int S_in = (l == 0) ? 1 : 32;
    const int x_in = (l == 0) ? 16 : 32;
    const float* Ain = (l == 0) ? adj  : A0;   // layers 1,2 update A in place
    float*       Aout = A0;
    const float* Xin = (l == 0) ? feat : ((l == 1) ? Xa : Xb);
    float*       Xout = (l == 0) ? Xa  : ((l == 1) ? Xb : (float*)d_out);

    void* const* pr = d_in + 2 + l * 24;   // per-layer params, pytree-sorted keys
    const float* A_bias = (const float*)pr[0];
    const float* A_l1 = (const float*)pr[1];
    const float* A_l2 = (const float*)pr[2];
    const float* A_l3 = (const float*)pr[3];
    const float* A_l4 = (const float*)pr[4];
    const float* A_l5 = (const float*)pr[5];
    const float* A_l6 = (const float*)pr[6];
    const float* A_l7 = (const float*)pr[7];
    const float* X1_bias = (const float*)pr[8];
    const float* X1_l1 = (const float*)pr[9];
    const float* X1_l2 = (const float*)pr[10];
    const float* X1_l3 = (const float*)pr[11];
    const float* X1_l4 = (const float*)pr[12];
    const float* X1_l5 = (const float*)pr[13];
    const float* X1_l6 = (const float*)pr[14];
    const float* X2_bias = (const float*)pr[15];
    const float* X2_l1 = (const float*)pr[16];
    const float* X2_l2 = (const float*)pr[17];
    const float* X2_l3 = (const float*)pr[18];
    const float* X2_l4 = (const float*)pr[19];
    const float* X2_l5 = (const float*)pr[20];
    const float* X2_l6 = (const float*)pr[21];
    const float* out_W = (const float*)pr[22];
    const float* out_b = (const float*)pr[23];

    if (S_in == 1) k_rowstats<1> <<<NB * NN, 32, 0, stream>>>(Ain, moc, diag);
    else           k_rowstats<32><<<NB * NN, 32, 0, stream>>>(Ain, moc, diag);

    k_means_consts<<<NB, 32, 0, stream>>>(moc, diag, Xin, x_in, S_in,
        A_bias, A_l2, A_l4, A_l7, X1_bias, X1_l2, X1_l5, X1_l6,
        X2_bias, X2_l2, X2_l5, X2_l6, baseB, c1, c2);

    k_rowmats<<<NB * 32, dim3(32, 8), 0, stream>>>(moc, diag, Xin, x_in, S_in,
        A_l3, A_l5, A_l6, X1_l1, X1_l3, X1_l4, X2_l1, X2_l3, X2_l4,
        c1, c2, rArr, X1t, X2tT);

    if (S_in == 1) k_At<1> <<<NB * NN * 16, 32, 0, stream>>>(Ain, A_l1, rArr, baseB, Aout);
    else           k_At<32><<<NB * NN * 16, 32, 0, stream>>>(Ain, A_l1, rArr, baseB, Aout);

    k_einsum_out<<<NB * 64, 128, 0, stream>>>(Aout, X2tT, X1t, out_W, out_b, Xout,
                                              (l < 2) ? 1 : 0);
  }
}